// DiceLoss_22840636080773
// MI455X (gfx1250) — compile-verified
//
#include <hip/hip_runtime.h>

// ---------------------------------------------------------------------------
// DiceLoss for MI455X (gfx1250, wave32).
//
// preds  : [8, 32, 512, 512] f32  (268 MB, streamed once -> memory bound,
//                                  ~12 us floor at 23.3 TB/s)
// targets: [8, 512, 512] int64 (values 0..31; we read only the low dword)
// out    : scalar f32
//
// Stage 1: per-pixel f32 softmax + three per-class reductions.
//   - lane = pixel => every per-channel global load is a coalesced 128B line
//   - S1 accumulated in 32 VGPRs per lane
//   - S2 / cnt accumulated via conflict-free transposed LDS float atomics
//     (addr = class*32 + lane  ->  bank = lane or lane+32, never conflicting)
//   - per-block partials (S1,S2,cnt)[32] written to d_ws
// Stage 2: column-sum of the 512x32 partial matrices with chained
//   V_WMMA_F32_16X16X4_F32 (A = ones). A=ones makes the result invariant to
//   the K-slot permutation inside B's layout; only lane->column mapping
//   (lanes 0-15 / 16-31 -> N=0..15) is relied upon. Full f32 precision.
// ---------------------------------------------------------------------------

#define NBLK 512
#define NTHR 256

static constexpr int  kC   = 32;
static constexpr int  kHW  = 512 * 512;            // 262144 = 2^18
static constexpr long long kNpix = 8LL * kHW;      // 2097152 pixels
static constexpr float kEps    = 1e-8f;
static constexpr float kSmooth = 1e-5f;
static constexpr float kLog2e  = 1.44269504088896340736f;

typedef float v2f __attribute__((ext_vector_type(2)));
typedef float v8f __attribute__((ext_vector_type(8)));

__global__ __launch_bounds__(NTHR) void dice_stage1(
    const float* __restrict__ preds,
    const int*   __restrict__ targ2,   // int64 targets viewed as int32 pairs
    float*       __restrict__ part)    // [3][NBLK][32]
{
    __shared__ float s1t[32 * 32];   // [class][lane-bin]
    __shared__ float s2t[32 * 32];
    __shared__ float cnt[32 * 32];

    const int tid  = threadIdx.x;
    const int lane = tid & 31;

    for (int i = tid; i < 32 * 32; i += NTHR) {
        s1t[i] = 0.f; s2t[i] = 0.f; cnt[i] = 0.f;
    }
    __syncthreads();

    float acc[kC];
#pragma unroll
    for (int c = 0; c < kC; ++c) acc[c] = 0.f;

    const long long T = (long long)NBLK * NTHR;    // 131072 threads, 16 px each
    for (long long n = (long long)blockIdx.x * NTHR + tid; n < kNpix; n += T) {
        const long long b = n >> 18;               // batch
        const long long p = n & (kHW - 1);         // pixel within batch
        const float* pp = preds + (b << 23) + p;   // 32*HW = 2^23

        // 32 coalesced, non-temporal channel loads (each byte read once).
        float x[kC];
#pragma unroll
        for (int c = 0; c < kC; ++c)
            x[c] = __builtin_nontemporal_load(pp + ((long long)c << 18));

        // softmax in f32
        float m = x[0];
#pragma unroll
        for (int c = 1; c < kC; ++c) m = fmaxf(m, x[c]);
        float s = 0.f;
#pragma unroll
        for (int c = 0; c < kC; ++c) {
            x[c] = __builtin_amdgcn_exp2f((x[c] - m) * kLog2e);  // v_exp_f32
            s += x[c];
        }
        const float inv = __builtin_amdgcn_rcpf(s);              // v_rcp_f32

        // S1 accumulation in registers
#pragma unroll
        for (int c = 0; c < kC; ++c) acc[c] = fmaf(x[c], inv, acc[c]);

        // P at labeled class via in-register select chain (no re-load)
        const int t = targ2[2 * n];                 // low dword of int64 label
        float et = x[0];
#pragma unroll
        for (int c = 1; c < kC; ++c) et = (t == c) ? x[c] : et;

        // conflict-free transposed LDS atomic bins
        atomicAdd(&s2t[t * 32 + lane], et * inv);
        atomicAdd(&cnt[t * 32 + lane], 1.0f);
    }

    // drain per-lane S1 accumulators (conflict-free banks)
#pragma unroll
    for (int c = 0; c < kC; ++c) atomicAdd(&s1t[c * 32 + lane], acc[c]);
    __syncthreads();

    // per-block reduction of the 32 lane-bins -> one partial per class/array
    if (tid < 96) {
        const int a = tid >> 5, c = tid & 31;
        const float* src = (a == 0) ? s1t : (a == 1) ? s2t : cnt;
        float r = 0.f;
#pragma unroll
        for (int j = 0; j < 32; ++j) r += src[c * 32 + j];
        part[(size_t)a * NBLK * 32 + (size_t)blockIdx.x * 32 + c] = r;
    }
}

__global__ __launch_bounds__(NTHR) void dice_stage2(
    const float* __restrict__ part,    // [3][NBLK][32]
    float*       __restrict__ out)
{
    __shared__ float fin[96];
    __shared__ float red[32];

    const int tid  = threadIdx.x;
    const int wave = tid >> 5;
    const int lane = tid & 31;

    // 6 waves: (array 0..2) x (column group 0..1). Branch is wave-uniform,
    // so EXEC is all-ones inside the WMMA region (required by the ISA).
    if (wave < 6) {
        const int a = wave >> 1;
        const int g = wave & 1;
        const float* src = part + (size_t)a * NBLK * 32 + g * 16;

        const int n    = lane & 15;
        const int half = lane >> 4;

        v2f ones; ones.x = 1.f; ones.y = 1.f;     // A = ones(16x4)
        v8f d = {};                                // f32 accumulator

        // colsum(512 x 16) via 128 chained f32 WMMAs: D += ones x B(4x16)
        for (int blk = 0; blk < NBLK; blk += 4) {
            v2f bm;
            bm.x = src[(size_t)(blk + 2 * half    ) * 32 + n];
            bm.y = src[(size_t)(blk + 2 * half + 1) * 32 + n];
            d = __builtin_amdgcn_wmma_f32_16x16x4_f32(
                    false, ones, false, bm, (short)0, d, false, false);
        }
        // D row 0: VGPR0, lanes 0-15 hold N = lane  -> column sums
        if (lane < 16) fin[a * 32 + g * 16 + lane] = d[0];
    }
    __syncthreads();

    if (tid < 32) {
        const float S1 = fin[tid], S2 = fin[32 + tid], CT = fin[64 + tid];
        const float TP = (1.f - kEps) * S2 + kEps * S1;
        const float FP = S1 - TP;
        const float FN = (1.f - kEps) * CT + kEps * (float)kNpix - TP;
        float alpha = FP / (FP + FN + kSmooth);
        alpha = fminf(fmaxf(alpha, 0.2f), 0.8f);
        const float beta = 1.f - alpha;
        const float dice = TP / (TP + alpha * FP + beta * FN + kSmooth);
        red[tid] = (1.f - dice) * (1.f / (float)kC);
    }
    __syncthreads();

    if (tid == 0) {
        float s = 0.f;
#pragma unroll
        for (int j = 0; j < 32; ++j) s += red[j];
        out[0] = s;
    }
}

extern "C" void kernel_launch(void* const* d_in, const int* in_sizes, int n_in,
                              void* d_out, int out_size, void* d_ws, size_t ws_size,
                              hipStream_t stream) {
    const float* preds = (const float*)d_in[0];
    const int*   targ2 = (const int*)d_in[1];   // int64 labels, int32 view
    float*       part  = (float*)d_ws;          // needs 3*NBLK*32*4 = 192 KiB

    dice_stage1<<<NBLK, NTHR, 0, stream>>>(preds, targ2, part);
    dice_stage2<<<1, NTHR, 0, stream>>>(part, (float*)d_out);
}